// MeshGraphEncoder_4698694221866
// MI455X (gfx1250) — compile-verified
//
#include <hip/hip_runtime.h>
#include <cstddef>

#define N_SRC 524288
#define N_DST 40962
#define N_EDGE 524288
#define DIM 128

typedef __attribute__((ext_vector_type(2))) float v2f;
typedef __attribute__((ext_vector_type(8))) float v8f;

// D = A(16x4 f32) * B(4x16 f32) + C(16x16 f32), wave32 WMMA.
__device__ __forceinline__ v8f wmma4(v2f a, v2f b, v8f c) {
  return __builtin_amdgcn_wmma_f32_16x16x4_f32(
      /*neg_a=*/false, a, /*neg_b=*/false, b,
      /*c_mod=*/(short)0, c, /*reuse_a=*/false, /*reuse_b=*/false);
}

// One wave computes a 32x16 output tile (two 16x16 M-tiles sharing the B
// fragment) for output columns [n0, n0+16).  X is in LDS (row stride LDX,
// padded so stride % 64 == 4 words -> conflict-free A loads).  W is a
// row-major [KDIM x 128] weight matrix in global memory.
template <int KDIM, int LDX>
__device__ __forceinline__ void gemm_pair(const float* __restrict__ W,
                                          const float* __restrict__ X,
                                          int n0, int l16, int half,
                                          v8f& acc0, v8f& acc1) {
#pragma unroll 4
  for (int k0 = 0; k0 < KDIM; k0 += 4) {
    const int kr = k0 + 2 * half;  // ISA A/B layout: lanes 16-31 hold K+2,K+3
    v2f b;
    b.x = W[(size_t)(kr + 0) * DIM + n0 + l16];
    b.y = W[(size_t)(kr + 1) * DIM + n0 + l16];
    const v2f a0 = *(const v2f*)(X + l16 * LDX + kr);          // rows 0..15
    const v2f a1 = *(const v2f*)(X + (l16 + 16) * LDX + kr);   // rows 16..31
    acc0 = wmma4(a0, b, acc0);
    acc1 = wmma4(a1, b, acc1);
  }
}

__device__ __forceinline__ float silu(float x) {
  return x / (1.0f + __expf(-x));
}

// ---------------------------------------------------------------------------
// Edge MLP: x = [efeat | grid[src] | mesh[dst]] (384) -> SiLU(xW1+b1) (128)
// -> LN(hW2+b2)*g+b -> atomic scatter-add into agg[dst].
// ---------------------------------------------------------------------------
__global__ __launch_bounds__(256, 2) void edge_mlp_scatter(
    const float* __restrict__ efeat, const float* __restrict__ grid,
    const float* __restrict__ mesh, const int* __restrict__ src_idx,
    const int* __restrict__ dst_idx, const float* __restrict__ W1,
    const float* __restrict__ b1, const float* __restrict__ W2,
    const float* __restrict__ b2, const float* __restrict__ lng,
    const float* __restrict__ lnb, float* __restrict__ agg) {
  extern __shared__ float smem[];
  __shared__ int sidx[32];
  __shared__ int didx[32];
  float* X = smem;              // 32 x 388
  float* H = smem + 32 * 388;   // 32 x 132
  float* Y = smem;              // aliases X (X dead after layer-1 barrier)
  const int tid = threadIdx.x;
  const int e0 = blockIdx.x * 32;
  if (tid < 32) {
    sidx[tid] = src_idx[e0 + tid];
    didx[tid] = dst_idx[e0 + tid];
  }
  __syncthreads();
  // Stage concat input rows: 32 rows x 96 float4.
  for (int i = tid; i < 32 * 96; i += 256) {
    const int r = i / 96;
    const int c = (i % 96) * 4;
    const float* p;
    if (c < 128)      p = efeat + (size_t)(e0 + r) * DIM + c;
    else if (c < 256) p = grid + (size_t)sidx[r] * DIM + (c - 128);
    else              p = mesh + (size_t)didx[r] * DIM + (c - 256);
    *(float4*)(X + r * 388 + c) = *(const float4*)p;
  }
  __syncthreads();
  const int lane = tid & 31;
  const int l16 = lane & 15;
  const int half = lane >> 4;
  const int n0 = (tid >> 5) * 16;
  v8f acc0 = {}, acc1 = {};
  gemm_pair<384, 388>(W1, X, n0, l16, half, acc0, acc1);
  {
    const float bb = b1[n0 + l16];
#pragma unroll
    for (int r = 0; r < 8; ++r) {
      H[(r + 8 * half) * 132 + n0 + l16] = silu(acc0[r] + bb);
      H[(16 + r + 8 * half) * 132 + n0 + l16] = silu(acc1[r] + bb);
    }
  }
  __syncthreads();
  v8f c0 = {}, c1 = {};
  gemm_pair<128, 132>(W2, H, n0, l16, half, c0, c1);
  {
    const float bb = b2[n0 + l16];
#pragma unroll
    for (int r = 0; r < 8; ++r) {
      Y[(r + 8 * half) * 132 + n0 + l16] = c0[r] + bb;
      Y[(16 + r + 8 * half) * 132 + n0 + l16] = c1[r] + bb;
    }
  }
  __syncthreads();
  // LayerNorm over 128 cols (8 lanes x 16 cols per row) + atomic scatter.
  const int row = tid >> 3;
  const int cg = tid & 7;
  const float* y = Y + row * 132 + cg * 16;
  float vals[16];
  float s = 0.f, ss = 0.f;
#pragma unroll
  for (int c = 0; c < 16; ++c) {
    const float v = y[c];
    vals[c] = v;
    s += v;
    ss += v * v;
  }
  s += __shfl_xor(s, 1); ss += __shfl_xor(ss, 1);
  s += __shfl_xor(s, 2); ss += __shfl_xor(ss, 2);
  s += __shfl_xor(s, 4); ss += __shfl_xor(ss, 4);
  const float m = s * (1.0f / 128.0f);
  const float rs = rsqrtf(ss * (1.0f / 128.0f) - m * m + 1e-5f);
  float* dp = agg + (size_t)didx[row] * DIM + cg * 16;
#pragma unroll
  for (int c = 0; c < 16; ++c) {
    const int col = cg * 16 + c;
    const float o = (vals[c] - m) * rs * lng[col] + lnb[col];
    __hip_atomic_fetch_add(dp + c, o, __ATOMIC_RELAXED, __HIP_MEMORY_SCOPE_AGENT);
  }
}

// ---------------------------------------------------------------------------
// Dst node MLP: x = [agg | mesh] (256) -> 128 -> LN -> mesh + out.  Guarded
// tail (N_DST = 40962 is not a multiple of 32; padded rows compute garbage
// but never load OOB and never store).
// ---------------------------------------------------------------------------
__global__ __launch_bounds__(256, 2) void dst_node_mlp(
    const float* __restrict__ agg, const float* __restrict__ mesh,
    const float* __restrict__ W1, const float* __restrict__ b1,
    const float* __restrict__ W2, const float* __restrict__ b2,
    const float* __restrict__ lng, const float* __restrict__ lnb,
    float* __restrict__ out) {
  extern __shared__ float smem[];
  float* X = smem;              // 32 x 260
  float* H = smem + 32 * 260;   // 32 x 132
  float* Y = smem;              // aliases X
  const int tid = threadIdx.x;
  const int r0 = blockIdx.x * 32;
  for (int i = tid; i < 32 * 64; i += 256) {
    const int r = i >> 6;
    const int c = (i & 63) * 4;
    const int gr = r0 + r;
    float4 v = make_float4(0.f, 0.f, 0.f, 0.f);
    if (gr < N_DST) {
      const float* p = (c < 128) ? (agg + (size_t)gr * DIM + c)
                                 : (mesh + (size_t)gr * DIM + (c - 128));
      v = *(const float4*)p;
    }
    *(float4*)(X + r * 260 + c) = v;
  }
  __syncthreads();
  const int lane = tid & 31;
  const int l16 = lane & 15;
  const int half = lane >> 4;
  const int n0 = (tid >> 5) * 16;
  v8f acc0 = {}, acc1 = {};
  gemm_pair<256, 260>(W1, X, n0, l16, half, acc0, acc1);
  {
    const float bb = b1[n0 + l16];
#pragma unroll
    for (int r = 0; r < 8; ++r) {
      H[(r + 8 * half) * 132 + n0 + l16] = silu(acc0[r] + bb);
      H[(16 + r + 8 * half) * 132 + n0 + l16] = silu(acc1[r] + bb);
    }
  }
  __syncthreads();
  v8f c0 = {}, c1 = {};
  gemm_pair<128, 132>(W2, H, n0, l16, half, c0, c1);
  {
    const float bb = b2[n0 + l16];
#pragma unroll
    for (int r = 0; r < 8; ++r) {
      Y[(r + 8 * half) * 132 + n0 + l16] = c0[r] + bb;
      Y[(16 + r + 8 * half) * 132 + n0 + l16] = c1[r] + bb;
    }
  }
  __syncthreads();
  const int row = tid >> 3;
  const int cg = tid & 7;
  const float* y = Y + row * 132 + cg * 16;
  float vals[16];
  float s = 0.f, ss = 0.f;
#pragma unroll
  for (int c = 0; c < 16; ++c) {
    const float v = y[c];
    vals[c] = v;
    s += v;
    ss += v * v;
  }
  s += __shfl_xor(s, 1); ss += __shfl_xor(ss, 1);
  s += __shfl_xor(s, 2); ss += __shfl_xor(ss, 2);
  s += __shfl_xor(s, 4); ss += __shfl_xor(ss, 4);
  const float m = s * (1.0f / 128.0f);
  const float rs = rsqrtf(ss * (1.0f / 128.0f) - m * m + 1e-5f);
  const int gr = r0 + row;
  if (gr < N_DST) {
#pragma unroll
    for (int c = 0; c < 16; ++c) {
      const int col = cg * 16 + c;
      out[(size_t)gr * DIM + col] =
          mesh[(size_t)gr * DIM + col] +
          (vals[c] - m) * rs * lng[col] + lnb[col];
    }
  }
}

// ---------------------------------------------------------------------------
// Src node MLP: grid (128) -> 128 -> LN -> grid + out.  N_SRC % 32 == 0.
// ---------------------------------------------------------------------------
__global__ __launch_bounds__(256, 2) void src_node_mlp(
    const float* __restrict__ x_in, const float* __restrict__ W1,
    const float* __restrict__ b1, const float* __restrict__ W2,
    const float* __restrict__ b2, const float* __restrict__ lng,
    const float* __restrict__ lnb, float* __restrict__ out) {
  extern __shared__ float smem[];
  float* X = smem;              // 32 x 132
  float* H = smem + 32 * 132;   // 32 x 132
  float* Y = smem;              // aliases X
  const int tid = threadIdx.x;
  const int r0 = blockIdx.x * 32;
  for (int i = tid; i < 32 * 32; i += 256) {
    const int r = i >> 5;
    const int c = (i & 31) * 4;
    *(float4*)(X + r * 132 + c) =
        *(const float4*)(x_in + (size_t)(r0 + r) * DIM + c);
  }
  __syncthreads();
  const int lane = tid & 31;
  const int l16 = lane & 15;
  const int half = lane >> 4;
  const int n0 = (tid >> 5) * 16;
  v8f acc0 = {}, acc1 = {};
  gemm_pair<128, 132>(W1, X, n0, l16, half, acc0, acc1);
  {
    const float bb = b1[n0 + l16];
#pragma unroll
    for (int r = 0; r < 8; ++r) {
      H[(r + 8 * half) * 132 + n0 + l16] = silu(acc0[r] + bb);
      H[(16 + r + 8 * half) * 132 + n0 + l16] = silu(acc1[r] + bb);
    }
  }
  __syncthreads();
  v8f c0 = {}, c1 = {};
  gemm_pair<128, 132>(W2, H, n0, l16, half, c0, c1);
  {
    const float bb = b2[n0 + l16];
#pragma unroll
    for (int r = 0; r < 8; ++r) {
      Y[(r + 8 * half) * 132 + n0 + l16] = c0[r] + bb;
      Y[(16 + r + 8 * half) * 132 + n0 + l16] = c1[r] + bb;
    }
  }
  __syncthreads();
  const int row = tid >> 3;
  const int cg = tid & 7;
  const float* y = Y + row * 132 + cg * 16;
  float vals[16];
  float s = 0.f, ss = 0.f;
#pragma unroll
  for (int c = 0; c < 16; ++c) {
    const float v = y[c];
    vals[c] = v;
    s += v;
    ss += v * v;
  }
  s += __shfl_xor(s, 1); ss += __shfl_xor(ss, 1);
  s += __shfl_xor(s, 2); ss += __shfl_xor(ss, 2);
  s += __shfl_xor(s, 4); ss += __shfl_xor(ss, 4);
  const float m = s * (1.0f / 128.0f);
  const float rs = rsqrtf(ss * (1.0f / 128.0f) - m * m + 1e-5f);
  const int gr = r0 + row;
#pragma unroll
  for (int c = 0; c < 16; ++c) {
    const int col = cg * 16 + c;
    out[(size_t)gr * DIM + col] =
        x_in[(size_t)gr * DIM + col] + (vals[c] - m) * rs * lng[col] + lnb[col];
  }
}

__global__ void zero_f4(float4* __restrict__ p, int n4) {
  const int i = blockIdx.x * blockDim.x + threadIdx.x;
  if (i < n4) p[i] = make_float4(0.f, 0.f, 0.f, 0.f);
}

extern "C" void kernel_launch(void* const* d_in, const int* in_sizes, int n_in,
                              void* d_out, int out_size, void* d_ws,
                              size_t ws_size, hipStream_t stream) {
  (void)in_sizes; (void)n_in; (void)out_size; (void)ws_size;
  const float* efeat = (const float*)d_in[0];
  const float* grid  = (const float*)d_in[1];
  const float* mesh  = (const float*)d_in[2];
  const int* src_idx = (const int*)d_in[3];
  const int* dst_idx = (const int*)d_in[4];
  const float* eW1 = (const float*)d_in[5];
  const float* eb1 = (const float*)d_in[6];
  const float* eW2 = (const float*)d_in[7];
  const float* eb2 = (const float*)d_in[8];
  const float* elng = (const float*)d_in[9];
  const float* elnb = (const float*)d_in[10];
  const float* sW1 = (const float*)d_in[11];
  const float* sb1 = (const float*)d_in[12];
  const float* sW2 = (const float*)d_in[13];
  const float* sb2 = (const float*)d_in[14];
  const float* slng = (const float*)d_in[15];
  const float* slnb = (const float*)d_in[16];
  const float* dW1 = (const float*)d_in[17];
  const float* db1 = (const float*)d_in[18];
  const float* dW2 = (const float*)d_in[19];
  const float* db2 = (const float*)d_in[20];
  const float* dlng = (const float*)d_in[21];
  const float* dlnb = (const float*)d_in[22];

  float* grid_out = (float*)d_out;                       // [N_SRC, 128]
  float* mesh_out = grid_out + (size_t)N_SRC * DIM;      // [N_DST, 128]
  float* agg = (float*)d_ws;                             // [N_DST, 128]

  const size_t EDGE_SMEM = (size_t)(32 * 388 + 32 * 132) * sizeof(float);
  const size_t DST_SMEM  = (size_t)(32 * 260 + 32 * 132) * sizeof(float);
  const size_t SRC_SMEM  = (size_t)(32 * 132 + 32 * 132) * sizeof(float);

  const int n4 = N_DST * DIM / 4;
  zero_f4<<<(n4 + 255) / 256, 256, 0, stream>>>((float4*)agg, n4);

  edge_mlp_scatter<<<N_EDGE / 32, 256, EDGE_SMEM, stream>>>(
      efeat, grid, mesh, src_idx, dst_idx, eW1, eb1, eW2, eb2, elng, elnb, agg);

  dst_node_mlp<<<(N_DST + 31) / 32, 256, DST_SMEM, stream>>>(
      agg, mesh, dW1, db1, dW2, db2, dlng, dlnb, mesh_out);

  src_node_mlp<<<N_SRC / 32, 256, SRC_SMEM, stream>>>(
      grid, sW1, sb1, sW2, sb2, slng, slnb, grid_out);
}